// JKnetLayer_20667382628950
// MI455X (gfx1250) — compile-verified
//
#include <hip/hip_runtime.h>
#include <hip/hip_bf16.h>
#include <stdint.h>

#define DF    96      // feature width (floats)
#define HOPS  4
#define TILE  256     // edges staged per block
#define BLK   256     // threads per block (8 wave32)

// ---- CDNA5 async global->LDS copy (ASYNCcnt-tracked) --------------------
__device__ __forceinline__ void async_g2l_b32(void* lds_dst, const void* gptr) {
    uint32_t lds_off = (uint32_t)(uintptr_t)lds_dst;   // low 32 bits of LDS aperture addr = LDS byte offset
    uint64_t gaddr   = (uint64_t)(uintptr_t)gptr;
    asm volatile("global_load_async_to_lds_b32 %0, %1, off"
                 :
                 : "v"(lds_off), "v"(gaddr)
                 : "memory");
}
__device__ __forceinline__ void wait_async0() {
    asm volatile("s_wait_asynccnt 0x0" ::: "memory");
}

// ---- scatter: accum[dst] += e * feat[src] -------------------------------
// One block stages TILE edge tuples into LDS via async copies, packed as
// 16B-aligned {src,dst,w,pad} slots; each wave32 then owns 32 edges.  The
// per-edge tuple broadcast is a single wave-uniform ds_load_b128.  For each
// edge the whole wave cooperates: lane L handles feature columns
// {L, L+32, L+64} -> 3 coalesced b32 loads and 3 coalesced
// global_atomic_add_f32 into the L2-resident accumulator.
__global__ __launch_bounds__(BLK) void scatter_kernel(const int*   __restrict__ src,
                                                      const int*   __restrict__ dst,
                                                      const float* __restrict__ ew,
                                                      const float* __restrict__ feat,
                                                      int ld_feat,
                                                      float* __restrict__ accum,
                                                      int E) {
    __shared__ uint4 s_tup[TILE];   // .x=src  .y=dst  .z=bits(w)  .w=pad

    const int tid  = threadIdx.x;
    const int base = blockIdx.x * TILE;
    int n = E - base; if (n > TILE) n = TILE;

    if (tid < n) {
        async_g2l_b32(&s_tup[tid].x, src + base + tid);
        async_g2l_b32(&s_tup[tid].y, dst + base + tid);
        async_g2l_b32(&s_tup[tid].z, ew  + base + tid);
    }
    wait_async0();
    __syncthreads();

    const int wave = tid >> 5;
    const int lane = tid & 31;
    const int jbeg = wave * 32;
    int jend = jbeg + 32; if (jend > n) jend = n;

    for (int j = jbeg; j < jend; ++j) {
        const uint4 t = s_tup[j];                     // one ds_load_b128, wave-uniform
        const float w = __uint_as_float(t.z);
        const float* __restrict__ row = feat  + (size_t)t.x * ld_feat;
        float*       __restrict__ acc = accum + (size_t)t.y * DF;
#pragma unroll
        for (int k = 0; k < 3; ++k) {
            float v = row[lane + 32 * k] * w;
            unsafeAtomicAdd(acc + lane + 32 * k, v);  // hardware global_atomic_add_f32
        }
    }
}

// ---- per-edge weight: e = d[src]*d[dst] ---------------------------------
__global__ __launch_bounds__(BLK) void ew_kernel(const int* __restrict__ src,
                                                 const int* __restrict__ dst,
                                                 const float* __restrict__ d,
                                                 float* __restrict__ ew, int E) {
    int i = blockIdx.x * BLK + threadIdx.x;
    if (i < E) ew[i] = d[src[i]] * d[dst[i]];
}

// ---- zero the accumulator (float4-wide) ---------------------------------
__global__ __launch_bounds__(BLK) void zero_kernel(float4* __restrict__ p, int n4) {
    int i = blockIdx.x * BLK + threadIdx.x;
    if (i < n4) p[i] = make_float4(0.f, 0.f, 0.f, 0.f);
}

// ---- blend: out = r*accum + (1-r)*feat_in; reset accum ------------------
// Writes directly into the hop's slice of d_out; that slice is the input
// of the next hop (ld = 4*DF), so no extra feature buffers are needed.
__global__ __launch_bounds__(BLK) void blend_kernel(float*       __restrict__ accum,
                                                    const float* __restrict__ feat_in,
                                                    int ld_in,
                                                    float*       __restrict__ out,
                                                    int hop,
                                                    const float* __restrict__ lr,
                                                    int N) {
    const int i = blockIdx.x * BLK + threadIdx.x;
    const int total = N * (DF / 4);
    if (i >= total) return;

    const float r  = lr[hop];
    const float om = 1.0f - r;
    const int node = i / (DF / 4);
    const int c    = i - node * (DF / 4);

    const float4 a = *((const float4*)(accum + (size_t)i * 4));
    const float4 f = *((const float4*)(feat_in + (size_t)node * ld_in) + c);

    float4 o;
    o.x = r * a.x + om * f.x;
    o.y = r * a.y + om * f.y;
    o.z = r * a.z + om * f.z;
    o.w = r * a.w + om * f.w;

    *((float4*)(out + (size_t)node * (HOPS * DF) + hop * DF) + c) = o;
    *((float4*)(accum + (size_t)i * 4)) = make_float4(0.f, 0.f, 0.f, 0.f);
}

// -------------------------------------------------------------------------
extern "C" void kernel_launch(void* const* d_in, const int* in_sizes, int n_in,
                              void* d_out, int out_size, void* d_ws, size_t ws_size,
                              hipStream_t stream) {
    const float* h   = (const float*)d_in[0];
    const float* d   = (const float*)d_in[1];
    const int*   src = (const int*)d_in[2];
    const int*   dst = (const int*)d_in[3];
    const float* lr  = (const float*)d_in[4];
    float*       out = (float*)d_out;

    const int N = in_sizes[0] / DF;
    const int E = in_sizes[2];

    // workspace: e[E] | accum[N*DF]
    float* ew    = (float*)d_ws;
    float* accum = ew + E;          // E*4 bytes stays 16B-aligned for E=800000

    ew_kernel<<<(E + BLK - 1) / BLK, BLK, 0, stream>>>(src, dst, d, ew, E);

    const int n4 = N * (DF / 4);
    zero_kernel<<<(n4 + BLK - 1) / BLK, BLK, 0, stream>>>((float4*)accum, n4);

    const float* fin = h;
    int ld = DF;
    for (int hop = 0; hop < HOPS; ++hop) {
        scatter_kernel<<<(E + TILE - 1) / TILE, BLK, 0, stream>>>(
            src, dst, ew, fin, ld, accum, E);
        blend_kernel<<<(n4 + BLK - 1) / BLK, BLK, 0, stream>>>(
            accum, fin, ld, out, hop, lr, N);
        fin = out + hop * DF;   // this hop's slice feeds the next hop
        ld  = HOPS * DF;
    }
}